// VectorFNO2D_34643206210114
// MI455X (gfx1250) — compile-verified
//
#include <hip/hip_runtime.h>
#include <hip/hip_bf16.h>
#include <math.h>

// ---------------- constants ----------------
#define BSZ   8
#define S     256
#define NPIX  (S*S)
#define CW    32          // width (channels)
#define M1    12
#define M2    12
#define NL    4
#define WF    128
#define HP    288         // padded size (S + S/PADF)
#define KX24  24          // number of kept kx rows (0..11 and 276..287)
#define TN    32          // padded column count for T1 (24 used)

typedef float v2f __attribute__((ext_vector_type(2)));
typedef float v8f __attribute__((ext_vector_type(8)));

// ---------------- WMMA wrapper (fp32 16x16x4) ----------------
__device__ __forceinline__ v8f wmma_f32_16x16x4(v2f a, v2f b, v8f c) {
#if defined(__gfx1250__) && __has_builtin(__builtin_amdgcn_wmma_f32_16x16x4_f32)
  // 8 args: (neg_a, A, neg_b, B, c_mod, C, reuse_a, reuse_b)
  return __builtin_amdgcn_wmma_f32_16x16x4_f32(false, a, false, b, (short)0, c,
                                               false, false);
#else
  (void)a; (void)b;
  return c; // host pass / non-gfx1250 placeholder
#endif
}

__device__ __forceinline__ float gelu_exact(float v) {
  return 0.5f * v * (1.0f + erff(v * 0.7071067811865476f));
}

// ---------------- workspace layout (float offsets) ----------------
static constexpr size_t NH    = (size_t)BSZ * CW * HP * HP;           // 21,233,664
static constexpr size_t OFF_HA = 0;
static constexpr size_t OFF_HB = OFF_HA + NH;
static constexpr size_t OFF_T1 = OFF_HB + NH;                          // 73728*32
static constexpr size_t OFF_XF = OFF_T1 + (size_t)BSZ*CW*HP*TN;        // B*C*24*24
static constexpr size_t OFF_YF = OFF_XF + (size_t)BSZ*CW*KX24*M2*2;
static constexpr size_t OFF_Z  = OFF_YF + (size_t)BSZ*CW*KX24*M2*2;    // B*C*288*24
static constexpr size_t OFF_O2 = OFF_Z  + (size_t)BSZ*CW*HP*M2*2;
static constexpr size_t OFF_WN = OFF_O2 + (size_t)BSZ*2*NPIX;          // 288*32
static constexpr size_t OFF_GX = OFF_WN + (size_t)HP*TN;               // 288*24*2
static constexpr size_t OFF_GY = OFF_GX + (size_t)HP*KX24*2;           // 288*12*2

// ---------------- twiddle tables ----------------
__global__ void init_tables_k(float* __restrict__ Wn, float* __restrict__ Gx,
                              float* __restrict__ Gy) {
  const float TWO_PI = 6.2831853071795864769f;
  int t = blockIdx.x * 256 + threadIdx.x;
  // Wn: (288 x 32), col 2ky = cos, 2ky+1 = -sin  (e^{-i th} components)
  if (t < HP * TN) {
    int n = t / TN, col = t % TN;
    float v = 0.0f;
    if (col < 2 * M2) {
      int ky = col >> 1;
      int r = (ky * n) % HP;
      float th = TWO_PI * (float)r / (float)HP;
      v = (col & 1) ? -sinf(th) : cosf(th);
    }
    Wn[t] = v;
  }
  // Gx: (288 m x 24 kxidx) pairs (cos, sin) of +2*pi*kx*m/288
  if (t < HP * KX24) {
    int m = t / KX24, kxi = t % KX24;
    int kx = (kxi < M1) ? kxi : (HP - M1 + (kxi - M1));
    int r = (kx * m) % HP;
    float th = TWO_PI * (float)r / (float)HP;
    Gx[t * 2]     = cosf(th);
    Gx[t * 2 + 1] = sinf(th);
  }
  // Gy: (288 n x 12 ky) pairs scaled by c_ky/(H*W)
  if (t < HP * M2) {
    int n = t / M2, ky = t % M2;
    int r = (ky * n) % HP;
    float th = TWO_PI * (float)r / (float)HP;
    float s = ((ky == 0) ? 1.0f : 2.0f) / ((float)HP * (float)HP);
    Gy[t * 2]     = s * cosf(th);
    Gy[t * 2 + 1] = s * sinf(th);
  }
}

// ---------------- K0: gather + fc0 + pad ----------------
__global__ void init_h_k(const float* __restrict__ x, const int* __restrict__ d2v,
                         const float* __restrict__ fw, const float* __restrict__ fb,
                         float* __restrict__ h) {
  size_t p = (size_t)blockIdx.x * 256 + threadIdx.x;
  if (p >= NH) return;
  int n = (int)(p % HP);
  size_t q = p / HP;
  int m = (int)(q % HP); q /= HP;
  int c = (int)(q % CW);
  int b = (int)(q / CW);
  float val = 0.0f;
  if (m < S && n < S) {
    int idx = d2v[m * S + n];
    float xv = x[(size_t)b * NPIX + idx];
    float gi = (float)m * (1.0f / 255.0f);
    float gj = (float)n * (1.0f / 255.0f);
    val = xv * fw[c * 3 + 0] + gi * fw[c * 3 + 1] + gj * fw[c * 3 + 2] + fb[c];
  }
  h[p] = val;
}

// ---------------- forward DFT stage 1: big GEMM via WMMA ----------------
// T1 (73728 x 32) = h (73728 x 288) * Wn (288 x 32); 256 thr = 8 waves
__global__ void dft1_wmma_k(const float* __restrict__ h, const float* __restrict__ Wn,
                            float* __restrict__ T1) {
  __shared__ float Bs[HP * TN];
  for (int i = threadIdx.x; i < HP * TN; i += 256) Bs[i] = Wn[i];
  __syncthreads();
  int lane = threadIdx.x & 31, wave = threadIdx.x >> 5;
  size_t rowBase = (size_t)blockIdx.x * 128 + (size_t)wave * 16;
  const float* A = h + rowBase * HP;
  int mloc = lane & 15, halfk = lane >> 4;
  v8f c0 = {}, c1 = {};
  for (int kk = 0; kk < HP; kk += 4) {
    int k0 = kk + 2 * halfk;
    v2f a = *(const v2f*)(A + (size_t)mloc * HP + k0);
    v2f b0, b1;
    b0.x = Bs[k0 * TN + mloc];        b0.y = Bs[(k0 + 1) * TN + mloc];
    b1.x = Bs[k0 * TN + 16 + mloc];   b1.y = Bs[(k0 + 1) * TN + 16 + mloc];
    c0 = wmma_f32_16x16x4(a, b0, c0);
    c1 = wmma_f32_16x16x4(a, b1, c1);
  }
#pragma unroll
  for (int v = 0; v < 8; ++v) {
    int mr = v + ((lane >= 16) ? 8 : 0);
    T1[(rowBase + mr) * TN + mloc]      = c0[v];
    T1[(rowBase + mr) * TN + 16 + mloc] = c1[v];
  }
}

// ---------------- forward DFT stage 2 (along m) ----------------
// one block per (b,c); 288 thr = (kxidx 0..23, ky 0..11)
__global__ void dft2_k(const float* __restrict__ T1, const float* __restrict__ Gx,
                       float* __restrict__ XF) {
  __shared__ float s[HP * 24];
  size_t bc = blockIdx.x;
  for (int i = threadIdx.x; i < HP * 24; i += 288) {
    int mm = i / 24, j = i % 24;
    s[i] = T1[(bc * HP + mm) * TN + j];
  }
  __syncthreads();
  int kxi = threadIdx.x / M2, ky = threadIdx.x % M2;
  float ar = 0.0f, ai = 0.0f;
  for (int m = 0; m < HP; ++m) {
    float c = Gx[m * (KX24 * 2) + kxi * 2];
    float sn = Gx[m * (KX24 * 2) + kxi * 2 + 1];
    float tr = s[m * 24 + 2 * ky], ti = s[m * 24 + 2 * ky + 1];
    ar += tr * c + ti * sn;   // x * e^{-i th}
    ai += ti * c - tr * sn;
  }
  XF[(bc * KX24 + kxi) * 24 + ky * 2]     = ar;
  XF[(bc * KX24 + kxi) * 24 + ky * 2 + 1] = ai;
}

// ---------------- spectral mode mixing (complex 32x32 per mode) ----------------
// one block per mode (288); 256 thr = (b,o)
__global__ void modemix_k(const float* __restrict__ XF,
                          const float* __restrict__ w1r, const float* __restrict__ w1i,
                          const float* __restrict__ w2r, const float* __restrict__ w2i,
                          float* __restrict__ YF) {
  __shared__ float xs[BSZ * CW * 2];
  int mode = blockIdx.x;
  int kxi = mode / M2, ky = mode % M2;
  int t = threadIdx.x;
  {
    int bb = t >> 5, i = t & 31;
    size_t base = (((size_t)bb * CW + i) * KX24 + kxi) * 24 + ky * 2;
    xs[t * 2]     = XF[base];
    xs[t * 2 + 1] = XF[base + 1];
  }
  __syncthreads();
  int b = t >> 5, o = t & 31;
  const float* wr; const float* wi; int kxp;
  if (kxi < M1) { wr = w1r; wi = w1i; kxp = kxi; }
  else          { wr = w2r; wi = w2i; kxp = kxi - M1; }
  float yr = 0.0f, yi = 0.0f;
  for (int i = 0; i < CW; ++i) {
    float xr = xs[(b * CW + i) * 2], xi = xs[(b * CW + i) * 2 + 1];
    size_t wIdx = ((size_t)(i * CW + o) * M1 + kxp) * M2 + ky;
    float wre = wr[wIdx], wim = wi[wIdx];
    yr += xr * wre - xi * wim;
    yi += xr * wim + xi * wre;
  }
  size_t base = (((size_t)b * CW + o) * KX24 + kxi) * 24 + ky * 2;
  YF[base]     = yr;
  YF[base + 1] = yi;
}

// ---------------- inverse stage 1 (kx -> m) ----------------
// one block per (b,o); 288 thr = m
__global__ void inv1_k(const float* __restrict__ YF, const float* __restrict__ Gx,
                       float* __restrict__ Z) {
  __shared__ float ys[KX24 * M2 * 2];
  size_t bo = blockIdx.x;
  for (int i = threadIdx.x; i < KX24 * M2 * 2; i += 288) ys[i] = YF[bo * (KX24*M2*2) + i];
  __syncthreads();
  int m = threadIdx.x;
  float accr[M2], acci[M2];
#pragma unroll
  for (int ky = 0; ky < M2; ++ky) { accr[ky] = 0.0f; acci[ky] = 0.0f; }
  for (int kx = 0; kx < KX24; ++kx) {
    float c = Gx[m * (KX24 * 2) + kx * 2];
    float s = Gx[m * (KX24 * 2) + kx * 2 + 1];
#pragma unroll
    for (int ky = 0; ky < M2; ++ky) {
      float yrv = ys[kx * 24 + ky * 2], yiv = ys[kx * 24 + ky * 2 + 1];
      accr[ky] += yrv * c - yiv * s;   // y * e^{+i th}
      acci[ky] += yrv * s + yiv * c;
    }
  }
#pragma unroll
  for (int ky = 0; ky < M2; ++ky) {
    Z[(bo * HP + m) * 24 + 2 * ky]     = accr[ky];
    Z[(bo * HP + m) * 24 + 2 * ky + 1] = acci[ky];
  }
}

// ---------------- pointwise conv via WMMA: out = convW(32x32) * hslab + bias ----
// grid = B*HP*18 blocks; block = 64 thr (2 waves: o-halves)
__global__ void conv_wmma_k(const float* __restrict__ h, const float* __restrict__ cw,
                            const float* __restrict__ cb, float* __restrict__ out) {
  int bx = blockIdx.x;
  int ntile = bx % 18;
  int slab = bx / 18;              // b*HP + m
  int bidx = slab / HP, m = slab % HP;
  int lane = threadIdx.x & 31, wave = threadIdx.x >> 5;
  int m0 = wave * 16;              // output-channel base
  int mloc = lane & 15, halfk = lane >> 4;
  int ncol = ntile * 16 + mloc;
  const float* hslab = h + ((size_t)bidx * CW * HP + m) * HP;
  v8f acc = {};
  for (int kk = 0; kk < CW; kk += 4) {
    int k0 = kk + 2 * halfk;
    v2f a = *(const v2f*)(cw + (m0 + mloc) * CW + k0);
    v2f bf;
    bf.x = hslab[(size_t)k0 * (HP * HP) + ncol];
    bf.y = hslab[(size_t)(k0 + 1) * (HP * HP) + ncol];
    acc = wmma_f32_16x16x4(a, bf, acc);
  }
#pragma unroll
  for (int v = 0; v < 8; ++v) {
    int o = m0 + v + ((lane >= 16) ? 8 : 0);
    out[(((size_t)bidx * CW + o) * HP + m) * HP + ntile * 16 + mloc] = acc[v] + cb[o];
  }
}

// ---------------- inverse stage 2 + add + (gelu) ----------------
// one block per row (b,o,m); 288 thr = n
__global__ void inv2_add_k(const float* __restrict__ Z, const float* __restrict__ Gy,
                           float* __restrict__ hio, int apply_gelu) {
  __shared__ float zr[24];
  size_t row = blockIdx.x;
  if (threadIdx.x < 24) zr[threadIdx.x] = Z[row * 24 + threadIdx.x];
  __syncthreads();
  int n = threadIdx.x;
  float sp = 0.0f;
#pragma unroll
  for (int ky = 0; ky < M2; ++ky) {
    float gc = Gy[n * (M2 * 2) + ky * 2];
    float gs = Gy[n * (M2 * 2) + ky * 2 + 1];
    sp += zr[2 * ky] * gc - zr[2 * ky + 1] * gs;   // Re(z * e^{+i th}) scaled
  }
  float val = hio[row * HP + n] + sp;
  if (apply_gelu) val = gelu_exact(val);
  hio[row * HP + n] = val;
}

// ---------------- final MLP via WMMA (per (b,m<256) slab) ----------------
// grid = B*256, block = 256 thr, dyn LDS = 128*128 floats (64 KB)
__global__ void mlp_wmma_k(const float* __restrict__ h, const float* __restrict__ w1,
                           const float* __restrict__ b1, const float* __restrict__ w2,
                           const float* __restrict__ b2, float* __restrict__ O2) {
  extern __shared__ float smem[];  // O1 tile: [f*128 + n_local]
  int slab = blockIdx.x;
  int bidx = slab >> 8, m = slab & 255;
  const float* hslab = h + ((size_t)bidx * CW * HP + m) * HP;
  int lane = threadIdx.x & 31, wave = threadIdx.x >> 5;
  int mloc = lane & 15, halfk = lane >> 4;
  int f0 = wave * 16;
  for (int half = 0; half < 2; ++half) {
    int nbase = half * 128;
    for (int nt = 0; nt < 8; ++nt) {
      int nloc = nt * 16 + mloc;
      int ncol = nbase + nloc;
      v8f acc = {};
      for (int kk = 0; kk < CW; kk += 4) {
        int k0 = kk + 2 * halfk;
        v2f a = *(const v2f*)(w1 + (f0 + mloc) * CW + k0);
        v2f bf;
        bf.x = hslab[(size_t)k0 * (HP * HP) + ncol];
        bf.y = hslab[(size_t)(k0 + 1) * (HP * HP) + ncol];
        acc = wmma_f32_16x16x4(a, bf, acc);
      }
#pragma unroll
      for (int v = 0; v < 8; ++v) {
        int f = f0 + v + ((lane >= 16) ? 8 : 0);
        smem[f * 128 + nloc] = gelu_exact(acc[v] + b1[f]);
      }
    }
    __syncthreads();
    {
      int nloc = threadIdx.x & 127;
      int d = threadIdx.x >> 7;      // 0 or 1
      float a = 0.0f;
      for (int f = 0; f < WF; ++f) a += w2[d * WF + f] * smem[f * 128 + nloc];
      O2[((size_t)bidx * 2 + d) * NPIX + m * S + nbase + nloc] = a + b2[d];
    }
    __syncthreads();
  }
}

// ---------------- output permutation/scatter ----------------
__global__ void scatter_k(const float* __restrict__ O2, const int* __restrict__ v2d,
                          float* __restrict__ out) {
  size_t p = (size_t)blockIdx.x * 256 + threadIdx.x;   // b*NPIX + n
  int b = (int)(p >> 16);
  int n = (int)(p & (NPIX - 1));
  int sidx = v2d[n];
  out[((size_t)b << 17) + (size_t)n * 2]     = O2[((size_t)b * 2 + 0) * NPIX + sidx];
  out[((size_t)b << 17) + (size_t)n * 2 + 1] = O2[((size_t)b * 2 + 1) * NPIX + sidx];
}

// ---------------- launch ----------------
extern "C" void kernel_launch(void* const* d_in, const int* in_sizes, int n_in,
                              void* d_out, int out_size, void* d_ws, size_t ws_size,
                              hipStream_t stream) {
  (void)in_sizes; (void)n_in; (void)out_size; (void)ws_size;
  const float* x      = (const float*)d_in[0];
  const int*   d2v    = (const int*)d_in[1];
  const int*   v2d    = (const int*)d_in[2];
  const float* fc0_w  = (const float*)d_in[3];
  const float* fc0_b  = (const float*)d_in[4];
  const float* w1r    = (const float*)d_in[5];
  const float* w1i    = (const float*)d_in[6];
  const float* w2r    = (const float*)d_in[7];
  const float* w2i    = (const float*)d_in[8];
  const float* conv_w = (const float*)d_in[9];
  const float* conv_b = (const float*)d_in[10];
  const float* mlp_w1 = (const float*)d_in[11];
  const float* mlp_b1 = (const float*)d_in[12];
  const float* mlp_w2 = (const float*)d_in[13];
  const float* mlp_b2 = (const float*)d_in[14];

  float* W  = (float*)d_ws;
  float* hA = W + OFF_HA;
  float* hB = W + OFF_HB;
  float* T1 = W + OFF_T1;
  float* XF = W + OFF_XF;
  float* YF = W + OFF_YF;
  float* Zb = W + OFF_Z;
  float* O2 = W + OFF_O2;
  float* Wn = W + OFF_WN;
  float* Gx = W + OFF_GX;
  float* Gy = W + OFF_GY;

  init_tables_k<<<36, 256, 0, stream>>>(Wn, Gx, Gy);
  init_h_k<<<(int)(NH / 256), 256, 0, stream>>>(x, d2v, fc0_w, fc0_b, hA);

  const size_t specStride = (size_t)CW * CW * M1 * M2;   // per-layer spectral weights
  for (int l = 0; l < NL; ++l) {
    dft1_wmma_k<<<576, 256, 0, stream>>>(hA, Wn, T1);
    dft2_k<<<BSZ * CW, 288, 0, stream>>>(T1, Gx, XF);
    modemix_k<<<KX24 * M2, 256, 0, stream>>>(XF,
        w1r + l * specStride, w1i + l * specStride,
        w2r + l * specStride, w2i + l * specStride, YF);
    inv1_k<<<BSZ * CW, 288, 0, stream>>>(YF, Gx, Zb);
    conv_wmma_k<<<BSZ * HP * 18, 64, 0, stream>>>(hA, conv_w + l * CW * CW,
                                                  conv_b + l * CW, hB);
    inv2_add_k<<<BSZ * CW * HP, 288, 0, stream>>>(Zb, Gy, hB, (l != NL - 1) ? 1 : 0);
    float* tmp = hA; hA = hB; hB = tmp;
  }

  mlp_wmma_k<<<BSZ * S, 256, 128 * 128 * sizeof(float), stream>>>(
      hA, mlp_w1, mlp_b1, mlp_w2, mlp_b2, O2);
  scatter_k<<<BSZ * NPIX / 256, 256, 0, stream>>>(O2, v2d, (float*)d_out);
}